// BIMPM_57621281243632
// MI455X (gfx1250) — compile-verified
//
#include <hip/hip_runtime.h>
#include <hip/hip_bf16.h>

typedef float v2f __attribute__((ext_vector_type(2)));
typedef float v8f __attribute__((ext_vector_type(8)));

#define EPSF 1e-8f
#define NEG_BIG -3.402823466e38f

// Problem dims (fixed by setup_inputs)
static constexpr int B  = 16;
static constexpr int S  = 128;
static constexpr int T  = 128;
static constexpr int H  = 256;   // half hidden
static constexpr int H2 = 512;
static constexpr int L  = 16;
static constexpr int OC = 128;   // output channels = 8*L

// LDS B-tile row pitch (floats): 260 = 256 + 4 pad so lane n (row n) maps to
// bank (4n + k) % 64 -> conflict-free across the 16 lanes of a WMMA B read.
static constexpr int BP = 260;

// ---------------------------------------------------------------------------
// Plain row norms: n1[b,half,s] = ||q1_half[b,s,:]||, same for n2.
// grid = B*2*2 (src), block = 256 (8 waves x 16 rows each)
// ---------------------------------------------------------------------------
__global__ __launch_bounds__(256)
void k_norm(const float* __restrict__ q1, const float* __restrict__ q2,
            float* __restrict__ n1, float* __restrict__ n2) {
    int blk = blockIdx.x;
    int src  = blk & 1;
    int half = (blk >> 1) & 1;
    int b    = blk >> 2;
    const float* Q = (src ? q2 : q1) + (size_t)b * S * H2 + half * H;
    float* dst = (src ? n2 : n1) + ((size_t)b * 2 + half) * S;
    int wave = threadIdx.x >> 5, lane = threadIdx.x & 31;
    for (int s = wave; s < S; s += 8) {
        float acc = 0.f;
        for (int h = lane; h < H; h += 32) {
            float v = Q[(size_t)s * H2 + h];
            acc += v * v;
        }
        for (int off = 16; off; off >>= 1) acc += __shfl_xor(acc, off);
        if (lane == 0) dst[s] = sqrtf(acc);
    }
}

// ---------------------------------------------------------------------------
// Weighted norms for max_pool_matching:
//   wna[b,half,l,s] = ||q1_half[b,s,:] * W[2+half][l,:]||   (wnb: q2)
// grid = B*2*L*2, block = 256
// ---------------------------------------------------------------------------
__global__ __launch_bounds__(256)
void k_wnorm(const float* __restrict__ q1, const float* __restrict__ q2,
             const float* __restrict__ W,
             float* __restrict__ wna, float* __restrict__ wnb) {
    int blk = blockIdx.x;
    int src  = blk & 1;
    int l    = (blk >> 1) % L;
    int half = ((blk >> 1) / L) & 1;
    int b    = blk / (2 * L * 2);
    const float* Q  = (src ? q2 : q1) + (size_t)b * S * H2 + half * H;
    const float* Wl = W + ((size_t)(2 + half) * L + l) * H;
    float* dst = (src ? wnb : wna) + (((size_t)(b * 2 + half)) * L + l) * S;
    int wave = threadIdx.x >> 5, lane = threadIdx.x & 31;
    for (int s = wave; s < S; s += 8) {
        float acc = 0.f;
        for (int h = lane; h < H; h += 32) {
            float v = Q[(size_t)s * H2 + h] * Wl[h];
            acc += v * v;
        }
        for (int off = 16; off; off >>= 1) acc += __shfl_xor(acc, off);
        if (lane == 0) dst[s] = sqrtf(acc);
    }
}

// ---------------------------------------------------------------------------
// Attention: mol = q1 . q2^T via WMMA f32 16x16x4, then
//   att[s,t] = mol / div_small(n1[s]*n2[t]);   attsum[s] = sum_t att[s,t]
// One wave per block; each wave owns one 16-row s-tile, all 128 t columns.
// grid = B*2*8, block = 32
// ---------------------------------------------------------------------------
__global__ __launch_bounds__(32)
void k_attention(const float* __restrict__ q1, const float* __restrict__ q2,
                 const float* __restrict__ n1, const float* __restrict__ n2,
                 float* __restrict__ att, float* __restrict__ attsum) {
    int blk = blockIdx.x;
    int stile = blk % 8;
    int half  = (blk / 8) & 1;
    int b     = blk / 16;
    const float* Q1 = q1 + (size_t)b * S * H2 + half * H;
    const float* Q2 = q2 + (size_t)b * S * H2 + half * H;
    int lane = threadIdx.x;
    int n  = lane & 15;     // A: M index / B: N index within tile
    int hi = lane >> 4;     // K sub-block selector (K = 2*hi, 2*hi+1)
    int s0 = stile * 16;

    v8f acc[8];
    for (int i = 0; i < 8; i++) acc[i] = {};

    for (int k = 0; k < H; k += 4) {
        int kk = k + 2 * hi;
        v2f a;
        a.x = Q1[(size_t)(s0 + n) * H2 + kk];
        a.y = Q1[(size_t)(s0 + n) * H2 + kk + 1];
        for (int tt = 0; tt < 8; tt++) {
            v2f bv;
            bv.x = Q2[(size_t)(tt * 16 + n) * H2 + kk];
            bv.y = Q2[(size_t)(tt * 16 + n) * H2 + kk + 1];
            acc[tt] = __builtin_amdgcn_wmma_f32_16x16x4_f32(
                false, a, false, bv, (short)0, acc[tt], false, false);
        }
    }

    const float* N1 = n1 + ((size_t)b * 2 + half) * S;
    const float* N2 = n2 + ((size_t)b * 2 + half) * S;
    float* ATT = att + (size_t)(b * 2 + half) * S * T;
    for (int r = 0; r < 8; r++) {
        int s = s0 + r + 8 * hi;
        float n1s = N1[s];
        float sum = 0.f;
        for (int tt = 0; tt < 8; tt++) {
            int t = tt * 16 + n;
            float den = n1s * N2[t];
            float a = acc[tt][r] / (den > EPSF ? den : EPSF);
            ATT[(size_t)s * T + t] = a;
            sum += a;
        }
        for (int off = 1; off < 16; off <<= 1) sum += __shfl_xor(sum, off);
        if (n == 0) attsum[((size_t)b * 2 + half) * S + s] = sum;
    }
}

// ---------------------------------------------------------------------------
// mean_q2[s,h] = (sum_t att[s,t]*q2[t,h]) / div_small(attsum[s])
// WMMA GEMM: A = att (128x128), B[k=t][n=h] = q2[t][h], K=128, N=256.
// grid = B*2*8 (s-tiles), block = 32 (one wave)
// ---------------------------------------------------------------------------
__global__ __launch_bounds__(32)
void k_mean(const float* __restrict__ q2, const float* __restrict__ att,
            const float* __restrict__ attsum, float* __restrict__ meanq2) {
    int blk = blockIdx.x;
    int stile = blk % 8;
    int half  = (blk / 8) & 1;
    int b     = blk / 16;
    const float* ATT = att + (size_t)(b * 2 + half) * S * T;
    const float* Q2  = q2 + (size_t)b * S * H2 + half * H;
    float* M = meanq2 + (size_t)(b * 2 + half) * S * H;
    int lane = threadIdx.x;
    int n  = lane & 15;
    int hi = lane >> 4;
    int s0 = stile * 16;

    for (int nt = 0; nt < 16; nt++) {       // 16 n-tiles cover H=256
        v8f acc = {};
        int h0 = nt * 16;
        for (int k = 0; k < T; k += 4) {
            int kk = k + 2 * hi;
            v2f a;
            a.x = ATT[(size_t)(s0 + n) * T + kk];
            a.y = ATT[(size_t)(s0 + n) * T + kk + 1];
            v2f bv;
            bv.x = Q2[(size_t)kk * H2 + h0 + n];
            bv.y = Q2[(size_t)(kk + 1) * H2 + h0 + n];
            acc = __builtin_amdgcn_wmma_f32_16x16x4_f32(
                false, a, false, bv, (short)0, acc, false, false);
        }
        for (int r = 0; r < 8; r++) {
            int s = s0 + r + 8 * hi;
            float d = attsum[((size_t)b * 2 + half) * S + s];
            d = (d > EPSF ? d : EPSF);
            M[(size_t)s * H + h0 + n] = acc[r] / d;
        }
    }
}

// ---------------------------------------------------------------------------
// amax[s,h] = max_t att[s,t]*q2[t,h]   (VALU; coalesced over h)
// grid = B*2*S, block = 256 (= H threads)
// ---------------------------------------------------------------------------
__global__ __launch_bounds__(256)
void k_amax(const float* __restrict__ q2, const float* __restrict__ att,
            float* __restrict__ amax) {
    int blk = blockIdx.x;
    int s    = blk % S;
    int half = (blk / S) & 1;
    int b    = blk / (S * 2);
    const float* Q2  = q2 + (size_t)b * S * H2 + half * H;
    const float* ATT = att + (size_t)(b * 2 + half) * S * T + (size_t)s * T;
    float* dst = amax + (((size_t)(b * 2 + half)) * S + s) * H;
    int h = threadIdx.x;
    float m = NEG_BIG;
    for (int t = 0; t < T; t++)
        m = fmaxf(m, ATT[t] * Q2[(size_t)t * H2 + h]);
    dst[h] = m;
}

// ---------------------------------------------------------------------------
// max_pool_matching: per (b,half,l)
//   dots[s,t] = sum_h q1[s,h]*q2[t,h]*w_l[h]^2   (w^2 folded into A)
//   cos = dots / (max(wna,EPS)*max(wnb,EPS));  out = max over t
// B operand (q2 t-tile, 16x256) is staged in LDS once per workgroup per
// t-tile via GLOBAL_LOAD_ASYNC_TO_LDS_B128 (ASYNCcnt path), then consumed by
// all 4 waves x 2 s-tiles from LDS (padded pitch -> conflict-free).
// grid = B*2*L, block = 128 (4 waves; each wave owns 2 s-tiles)
// ---------------------------------------------------------------------------
__global__ __launch_bounds__(128)
void k_maxpool(const float* __restrict__ q1, const float* __restrict__ q2,
               const float* __restrict__ W,
               const float* __restrict__ wna, const float* __restrict__ wnb,
               float* __restrict__ out) {
    __shared__ float bsh[16 * BP];   // 16 rows x 260 floats (~16.6 KB)

    int blk = blockIdx.x;
    int l    = blk % L;
    int half = (blk / L) & 1;
    int b    = blk / (L * 2);
    const float* Q1 = q1 + (size_t)b * S * H2 + half * H;
    const float* Q2 = q2 + (size_t)b * S * H2 + half * H;
    const float* Wl = W + ((size_t)(2 + half) * L + l) * H;
    const float* NA = wna + (((size_t)(b * 2 + half)) * L + l) * S;
    const float* NB = wnb + (((size_t)(b * 2 + half)) * L + l) * S;
    int tid  = threadIdx.x;
    int wave = tid >> 5, lane = tid & 31;
    int n  = lane & 15;
    int hi = lane >> 4;

    // Generic LDS pointer low 32 bits == LDS byte offset (ISA 10.2 aperture rule)
    unsigned lds_base = (unsigned)(size_t)(void*)bsh;

    float mx[2][8];
    for (int st = 0; st < 2; st++)
        for (int r = 0; r < 8; r++) mx[st][r] = NEG_BIG;

    for (int tt = 0; tt < 8; tt++) {
        int t0 = tt * 16;

        // --- async fill of B tile: 16 rows x 1KB, 16B per lane per issue ---
        {
            int chunk = tid & 63;    // 16B chunk within a row (64 per row)
            int rsel  = tid >> 6;    // 0..1 (two rows per issue across block)
            for (int j = 0; j < 8; j++) {
                int row = j * 2 + rsel;
                const float* gp = Q2 + (size_t)(t0 + row) * H2 + chunk * 4;
                unsigned ds = lds_base + (unsigned)(row * (BP * 4) + chunk * 16);
                asm volatile("global_load_async_to_lds_b128 %0, %1, off"
                             :: "v"(ds), "v"(gp) : "memory");
            }
            // Each wave waits for its own async loads; the barrier below then
            // makes every wave's LDS writes visible to the whole workgroup.
            asm volatile("s_wait_asynccnt 0" ::: "memory");
        }
        __syncthreads();

        for (int st = 0; st < 2; st++) {
            int s0 = (wave * 2 + st) * 16;
            v8f acc = {};
            for (int k = 0; k < H; k += 4) {
                int kk = k + 2 * hi;
                float w0 = Wl[kk], w1 = Wl[kk + 1];
                v2f a;
                a.x = Q1[(size_t)(s0 + n) * H2 + kk] * (w0 * w0);
                a.y = Q1[(size_t)(s0 + n) * H2 + kk + 1] * (w1 * w1);
                v2f bv;                       // B[k][n] = q2[t0+n][k] from LDS
                bv.x = bsh[n * BP + kk];
                bv.y = bsh[n * BP + kk + 1];
                acc = __builtin_amdgcn_wmma_f32_16x16x4_f32(
                    false, a, false, bv, (short)0, acc, false, false);
            }
            for (int r = 0; r < 8; r++) {
                int s = s0 + r + 8 * hi;
                int t = t0 + n;
                float na = NA[s]; na = (na > EPSF ? na : EPSF);
                float nb = NB[t]; nb = (nb > EPSF ? nb : EPSF);
                mx[st][r] = fmaxf(mx[st][r], acc[r] / (na * nb));
            }
        }
        __syncthreads();   // protect LDS before next t-tile overwrites it
    }

    for (int st = 0; st < 2; st++) {
        int s0 = (wave * 2 + st) * 16;
        for (int r = 0; r < 8; r++) {
            float v = mx[st][r];
            for (int off = 1; off < 16; off <<= 1) v = fmaxf(v, __shfl_xor(v, off));
            if (n == 0) {
                int s = s0 + r + 8 * hi;
                // group: mx_fw -> 2, mx_bw -> 3
                out[((size_t)b * S + s) * OC + (2 + half) * 16 + l] = v;
            }
        }
    }
}

// ---------------------------------------------------------------------------
// full_matching variants: f (anchor), at (mean_q2), mat (amax)
//   out[s,l] = (sum q1*v*w^2) / (max(||q1 w||,EPS)*max(||v w||,EPS))
// grid = B*2*S, block = 256 (8 waves; each wave covers 2 l values per variant)
// ---------------------------------------------------------------------------
__global__ __launch_bounds__(256)
void k_full(const float* __restrict__ q1, const float* __restrict__ q2,
            const float* __restrict__ W,
            const float* __restrict__ meanq2, const float* __restrict__ amax,
            float* __restrict__ out) {
    int blk = blockIdx.x;
    int s    = blk % S;
    int half = (blk / S) & 1;
    int b    = blk / (S * 2);
    const float* Q1 = q1 + (size_t)b * S * H2 + half * H + (size_t)s * H2;
    int arow = (half == 0) ? (S - 1) : 0;   // fw: last token, bw: first token
    const float* ANC  = q2 + (size_t)b * S * H2 + half * H + (size_t)arow * H2;
    const float* MEAN = meanq2 + (((size_t)(b * 2 + half)) * S + s) * H;
    const float* AMX  = amax   + (((size_t)(b * 2 + half)) * S + s) * H;
    int wave = threadIdx.x >> 5, lane = threadIdx.x & 31;

    const float* Vs[3] = { ANC, MEAN, AMX };
    int wIdx[3] = { 0 + half, 4 + half, 6 + half };  // W row group
    int gIdx[3] = { 0 + half, 4 + half, 6 + half };  // output group

    for (int vi = 0; vi < 3; vi++) {
        const float* V = Vs[vi];
        for (int l = wave; l < L; l += 8) {
            const float* Wl = W + ((size_t)wIdx[vi] * L + l) * H;
            float dab = 0.f, daa = 0.f, dbb = 0.f;
            for (int h = lane; h < H; h += 32) {
                float w = Wl[h];
                float a = Q1[h] * w;
                float c = V[h] * w;
                dab += a * c; daa += a * a; dbb += c * c;
            }
            for (int off = 16; off; off >>= 1) {
                dab += __shfl_xor(dab, off);
                daa += __shfl_xor(daa, off);
                dbb += __shfl_xor(dbb, off);
            }
            if (lane == 0) {
                float na = sqrtf(daa); na = (na > EPSF ? na : EPSF);
                float nb = sqrtf(dbb); nb = (nb > EPSF ? nb : EPSF);
                out[((size_t)b * S + s) * OC + gIdx[vi] * 16 + l] = dab / (na * nb);
            }
        }
    }
}

// ---------------------------------------------------------------------------
extern "C" void kernel_launch(void* const* d_in, const int* in_sizes, int n_in,
                              void* d_out, int out_size, void* d_ws, size_t ws_size,
                              hipStream_t stream) {
    (void)in_sizes; (void)n_in; (void)out_size; (void)ws_size;
    const float* q1 = (const float*)d_in[0];
    const float* q2 = (const float*)d_in[1];
    const float* W  = (const float*)d_in[2];
    float* out = (float*)d_out;
    float* ws  = (float*)d_ws;

    // workspace layout (floats)
    float* n1     = ws;             // B*2*S            = 4096
    float* n2     = ws + 4096;      // 4096
    float* attsum = ws + 8192;      // 4096
    float* wna    = ws + 12288;     // B*2*L*S          = 65536
    float* wnb    = ws + 77824;     // 65536
    float* att    = ws + 143360;    // B*2*S*T          = 524288
    float* meanq2 = ws + 667648;    // B*2*S*H          = 1048576
    float* amax   = ws + 1716224;   // 1048576  (total ~11 MB)

    k_norm     <<<B * 2 * 2,     256, 0, stream>>>(q1, q2, n1, n2);
    k_wnorm    <<<B * 2 * L * 2, 256, 0, stream>>>(q1, q2, W, wna, wnb);
    k_attention<<<B * 2 * 8,     32,  0, stream>>>(q1, q2, n1, n2, att, attsum);
    k_mean     <<<B * 2 * 8,     32,  0, stream>>>(q2, att, attsum, meanq2);
    k_amax     <<<B * 2 * S,     256, 0, stream>>>(q2, att, amax);
    k_maxpool  <<<B * 2 * L,     128, 0, stream>>>(q1, q2, W, wna, wnb, out);
    k_full     <<<B * 2 * S,     256, 0, stream>>>(q1, q2, W, meanq2, amax, out);
}